// DTPTrajRec_4724464025663
// MI455X (gfx1250) — compile-verified
//
#include <hip/hip_runtime.h>
#include <hip/hip_bf16.h>
#include <stdint.h>

// Flip to 0 to fall back to plain b128 load + nontemporal b128 store.
#define ASYNC_COPY 1

namespace {
constexpr int kB = 64;
constexpr int kL = 256;
constexpr int kT = 512;
constexpr int kD = 512;                 // per-table embedding dim
constexpr int kROW = 2 * kD;            // 1024 floats (4 KB) per output row
constexpr int kBT = kB * kT;            // 32768 rows per tensor
constexpr long long SRC_OFF = 0;
constexpr long long PRE_OFF = (long long)kBT * kROW;
constexpr long long NXT_OFF = 2LL * kBT * kROW;
constexpr long long GPS_OFF = 3LL * kBT * kROW;
}  // namespace

typedef float f4 __attribute__((ext_vector_type(4)));

// ---------------------------------------------------------------------------
// Kernel 1: init best-j table to -1 (workspace is poisoned by the harness).
// ---------------------------------------------------------------------------
__global__ void dtp_init_best(int* __restrict__ best) {
  int i = blockIdx.x * blockDim.x + threadIdx.x;
  if (i < kBT) best[i] = -1;
}

// ---------------------------------------------------------------------------
// Kernel 2: last-write-wins scatter == atomicMax over source index j.
// ---------------------------------------------------------------------------
__global__ void dtp_scatter_best(const int* __restrict__ src_grid,
                                 int* __restrict__ best) {
  int i = blockIdx.x * blockDim.x + threadIdx.x;  // over B*L
  if (i >= kB * kL) return;
  int b = i / kL;
  int j = i - b * kL;
  int t = src_grid[i * 3 + 2];
  if (t >= 0 && t < kT) atomicMax(&best[b * kT + t], j);
}

// ---------------------------------------------------------------------------
// Kernel 3: one block = one 4 KB output row = concat(E1[x], E2[y]).
//   blocks [0, BT)      -> src_emb (+ new_gps side output)
//   blocks [BT, 2BT)    -> pre_emb
//   blocks [2BT, 3BT)   -> next_emb
// Copy path: per-lane 16 B via async global->LDS->global DMA (CDNA5 path),
// NT store policy on the 403 MB streamed output.
// ---------------------------------------------------------------------------
__global__ void dtp_emit_rows(const int* __restrict__ src_grid,
                              const float* __restrict__ src_gps,
                              const int* __restrict__ pre_grids,
                              const int* __restrict__ nxt_grids,
                              const float* __restrict__ E1,
                              const float* __restrict__ E2,
                              const int* __restrict__ best,
                              float* __restrict__ out) {
  const int r = blockIdx.x;      // 0 .. 3*BT-1
  const int tid = threadIdx.x;   // 0 .. 255

  int x, y;
  float* dst;
  if (r < kBT) {
    const int b = r / kT;
    const int t = r - b * kT;
    const int bj = best[r];
    const bool has = (bj >= 0);
    const int sj = has ? bj : 0;
    const int gbase = (b * kL + sj) * 3;
    x = has ? src_grid[gbase + 0] : 0;  // empty slot -> E1[0]/E2[0], per reference
    y = has ? src_grid[gbase + 1] : 0;
    if (tid == 0) {
      float* ng = out + GPS_OFF + (long long)r * 3;
      ng[0] = has ? src_gps[(b * kL + sj) * 2 + 0] : 0.0f;
      ng[1] = has ? src_gps[(b * kL + sj) * 2 + 1] : 0.0f;
      ng[2] = has ? (float)t : 0.0f;
    }
    dst = out + SRC_OFF + (long long)r * kROW;
  } else if (r < 2 * kBT) {
    const int rr = r - kBT;      // rr = t*B + b over [T, B]
    x = pre_grids[rr * 3 + 0];
    y = pre_grids[rr * 3 + 1];
    dst = out + PRE_OFF + (long long)rr * kROW;
  } else {
    const int rr = r - 2 * kBT;
    x = nxt_grids[rr * 3 + 0];
    y = nxt_grids[rr * 3 + 1];
    dst = out + NXT_OFF + (long long)rr * kROW;
  }

  // Each thread moves one 16-byte chunk: lanes 0..127 -> E1 half, 128..255 -> E2 half.
  const float* srcp;
  float* dstp;
  if (tid < 128) {
    srcp = E1 + (long long)x * kD + tid * 4;
    dstp = dst + tid * 4;
  } else {
    const int k = tid - 128;
    srcp = E2 + (long long)y * kD + k * 4;
    dstp = dst + kD + k * 4;
  }

#if ASYNC_COPY
  __shared__ float stage[256 * 4];  // 4 KB staging: one private 16 B slot per thread
  const unsigned lds_off = (unsigned)(uintptr_t)(&stage[tid * 4]);
  // global -> LDS (per-lane 16 B; one VGLOBAL async op per wave; ASYNCcnt++)
  asm volatile("global_load_async_to_lds_b128 %0, %1, off"
               :
               : "v"(lds_off), "v"(srcp)
               : "memory");
  // lane-private slots: only need this wave's async load complete, no barrier
  asm volatile("s_wait_asynccnt 0x0" ::: "memory");
  // LDS -> global, non-temporal (streamed 403 MB output; keep E-tables in L2)
  asm volatile("global_store_async_from_lds_b128 %0, %1, off th:TH_STORE_NT"
               :
               : "v"(dstp), "v"(lds_off)
               : "memory");
  // s_endpgm's implicit wait-idle drains the outstanding async store.
#else
  f4 v = *(const f4*)srcp;
  __builtin_nontemporal_store(v, (f4*)dstp);
#endif
}

// ---------------------------------------------------------------------------
// Host entry
// ---------------------------------------------------------------------------
extern "C" void kernel_launch(void* const* d_in, const int* in_sizes, int n_in,
                              void* d_out, int out_size, void* d_ws, size_t ws_size,
                              hipStream_t stream) {
  (void)in_sizes; (void)n_in; (void)out_size; (void)ws_size;
  const int*   src_grid = (const int*)d_in[0];    // [B, L, 3]
  const float* src_gps  = (const float*)d_in[1];  // [B, L, 2]
  const int*   pre      = (const int*)d_in[2];    // [T, B, 3]
  const int*   nxt      = (const int*)d_in[3];    // [T, B, 3]
  const float* E1       = (const float*)d_in[4];  // [513, 512]
  const float* E2       = (const float*)d_in[5];  // [513, 512]
  float* out = (float*)d_out;
  int* best  = (int*)d_ws;                        // B*T ints = 128 KB

  dtp_init_best<<<kBT / 256, 256, 0, stream>>>(best);
  dtp_scatter_best<<<(kB * kL) / 256, 256, 0, stream>>>(src_grid, best);
  dtp_emit_rows<<<3 * kBT, 256, 0, stream>>>(src_grid, src_gps, pre, nxt,
                                             E1, E2, best, out);
}